// ModulatedConv2d_87316685127812
// MI455X (gfx1250) — compile-verified
//
#include <hip/hip_runtime.h>

typedef __attribute__((ext_vector_type(2))) float v2f;
typedef __attribute__((ext_vector_type(8))) float v8f;

#define B_   16
#define IC_  512
#define OC_  512
#define STY_ 512
#define HH   32
#define WW   32
#define UP   64
#define OUTS 63

#define MOD_SCALE 0.04419417382415922f   /* 1/sqrt(512)  */
#define CONV_SCALE 0.014731391274719738f /* 1/sqrt(4608) */
#define CONV_SCALE2 (1.0f/4608.0f)
#define EPS_ 1e-8f

// ---------------- K1: s[b,ic] = style @ (modw*ms)^T + bias ----------------
__global__ __launch_bounds__(256) void k_style(const float* __restrict__ style,
                                               const float* __restrict__ modw,
                                               const float* __restrict__ modb,
                                               float* __restrict__ s) {
  int idx = blockIdx.x * 256 + threadIdx.x;          // B_*IC_ = 8192
  int b = idx / IC_, ic = idx % IC_;
  const float* st = style + b * STY_;
  const float* mw = modw + ic * STY_;
  float acc = 0.f;
  for (int k = 0; k < STY_; ++k) acc += st[k] * mw[k];
  s[idx] = acc * MOD_SCALE + modb[ic];
}

// ------- K2: xinT[b][h][w][ic] = input[b][ic][h][w] * s[b,ic]  (NHWC) -------
__global__ __launch_bounds__(256) void k_modin(const float* __restrict__ in,
                                               const float* __restrict__ s,
                                               float* __restrict__ xinT) {
  int idx = blockIdx.x * 256 + threadIdx.x;          // B_*IC_*HH*WW = 8388608
  int bic = idx >> 10;                               // b*IC_ + ic
  int hw  = idx & 1023;
  int b   = idx >> 19;
  int ic  = bic & (IC_ - 1);
  xinT[(((size_t)b << 10) + hw) * IC_ + ic] = in[idx] * s[bic];
}

// ------- K2b: wT[tap][oc][ic] = weight[oc][ic][tap]  (K-innermost) ----------
__global__ __launch_bounds__(256) void k_wtr(const float* __restrict__ w,
                                             float* __restrict__ wT) {
  int idx = blockIdx.x * 256 + threadIdx.x;          // 9*OC_*IC_ = 2359296
  int tap = idx / (OC_ * IC_);
  int rem = idx - tap * (OC_ * IC_);                 // oc*IC_ + ic
  wT[idx] = w[(size_t)rem * 9 + tap];
}

// ---------------- K3a: wsq[oc,ic] = sum_taps w^2 ----------------
__global__ __launch_bounds__(256) void k_wsq(const float* __restrict__ w,
                                             float* __restrict__ wsq) {
  int idx = blockIdx.x * 256 + threadIdx.x;          // OC_*IC_ = 262144
  const float* p = w + idx * 9;
  float a = 0.f;
#pragma unroll
  for (int i = 0; i < 9; ++i) a += p[i] * p[i];
  wsq[idx] = a;
}

// ---------------- K3b: demod[b,oc] ----------------
__global__ __launch_bounds__(256) void k_demod(const float* __restrict__ s,
                                               const float* __restrict__ wsq,
                                               float* __restrict__ demod) {
  int idx = blockIdx.x * 256 + threadIdx.x;          // B_*OC_ = 8192
  int b = idx / OC_, oc = idx % OC_;
  const float* sp = s + b * IC_;
  const float* wq = wsq + oc * IC_;
  float acc = 0.f;
  for (int ic = 0; ic < IC_; ++ic) { float sv = sp[ic]; acc += sv * sv * wq[ic]; }
  demod[idx] = rsqrtf(CONV_SCALE2 * acc + EPS_);
}

// ---------------- K4: implicit-GEMM transposed conv (WMMA f32 16x16x4) -------
// One wave: 32 OC (two 16-row A tiles) x one full 32-wide quadrant row (two
// 16-col B tiles) -> 4 accumulators; K-loop over IC in steps of 4.
// K-innermost layouts make every fragment pair a single b64 load.
template <int NRT, int NCT>
__device__ __forceinline__ void conv_quad(const float* __restrict__ xb,
                                          const float* __restrict__ wT,
                                          float* __restrict__ ob,
                                          int r, int py, int px,
                                          int oc0, int lane) {
  const int m     = lane & 15;   // A row (oc) == B col (n)
  const int khalf = lane >> 4;   // lanes 16-31 hold K=2,3
  constexpr int NT = NRT * NCT;

  int ihs[NRT], wkh[NRT];
  if constexpr (NRT == 1) { ihs[0] = r;     wkh[0] = 1; }
  else                    { ihs[0] = r - 1; wkh[0] = 2; ihs[1] = r; wkh[1] = 0; }
  int dxs[NCT], wkw[NCT];
  if constexpr (NCT == 1) { dxs[0] = 0;  wkw[0] = 1; }
  else                    { dxs[0] = -1; wkw[0] = 2; dxs[1] = 0; wkw[1] = 0; }

  const float* aptr[NT];
  const float* bptr[NT][2];
  float        msk[NT][2];
#pragma unroll
  for (int ti = 0; ti < NRT; ++ti) {
#pragma unroll
    for (int tj = 0; tj < NCT; ++tj) {
      const int t = ti * NCT + tj;
      const int tap = wkh[ti] * 3 + wkw[tj];
      aptr[t] = wT + ((size_t)tap * OC_ + oc0 + m) * IC_ + khalf * 2;
      const bool rv = (ihs[ti] >= 0) & (ihs[ti] < HH);
      const int  ih = rv ? ihs[ti] : 0;
#pragma unroll
      for (int h = 0; h < 2; ++h) {
        const int  iw = h * 16 + m + dxs[tj];
        const bool v  = rv & (iw >= 0) & (iw < WW);
        msk[t][h]  = v ? 1.f : 0.f;
        bptr[t][h] = xb + ((size_t)ih * WW + (v ? iw : 0)) * IC_ + khalf * 2;
      }
    }
  }

  v8f acc[2][2] = {{{}, {}}, {{}, {}}};
#pragma unroll 2
  for (int ic0 = 0; ic0 < IC_; ic0 += 4) {
#pragma unroll
    for (int t = 0; t < NT; ++t) {
      v2f a0 = *(const v2f*)(aptr[t] + ic0);             // oc tile 0, one b64
      v2f a1 = *(const v2f*)(aptr[t] + 16 * IC_ + ic0);  // oc tile 1
      v2f b0 = *(const v2f*)(bptr[t][0] + ic0);          // col tile 0
      v2f b1 = *(const v2f*)(bptr[t][1] + ic0);          // col tile 1
      b0 *= msk[t][0];
      b1 *= msk[t][1];
      acc[0][0] = __builtin_amdgcn_wmma_f32_16x16x4_f32(
          false, a0, false, b0, (short)0, acc[0][0], false, false);
      acc[0][1] = __builtin_amdgcn_wmma_f32_16x16x4_f32(
          false, a0, false, b1, (short)0, acc[0][1], false, false);
      acc[1][0] = __builtin_amdgcn_wmma_f32_16x16x4_f32(
          false, a1, false, b0, (short)0, acc[1][0], false, false);
      acc[1][1] = __builtin_amdgcn_wmma_f32_16x16x4_f32(
          false, a1, false, b1, (short)0, acc[1][1], false, false);
    }
  }

  // D layout: lane holds col n=m, rows khalf*8 + i in VGPR i.
  const int y = 2 * r + py;
#pragma unroll
  for (int o = 0; o < 2; ++o) {
    float* opb = ob + (((size_t)(oc0 + o * 16 + khalf * 8)) * UP + y) * UP;
#pragma unroll
    for (int h = 0; h < 2; ++h) {
      const int x = 2 * (h * 16 + m) + px;
#pragma unroll
      for (int i = 0; i < 8; ++i) opb[(size_t)i * UP * UP + x] = acc[o][h][i];
    }
  }
}

__global__ __launch_bounds__(256) void k_conv(const float* __restrict__ xinT,
                                              const float* __restrict__ wT,
                                              float* __restrict__ outup) {
  const int lane = threadIdx.x & 31;
  const int wv   = threadIdx.x >> 5;
  int bi = blockIdx.x;                 // 4096 blocks, scalar decode
  const int ocB  = bi & 1;  bi >>= 1;  // 2 groups of 8 waves (32 OC each)
  const int r    = bi & 31; bi >>= 5;  // 32 sub-grid rows
  const int quad = bi & 3;  bi >>= 2;  // 4 parity quadrants (scalar)
  const int b    = bi;                 // 16 batches
  const int oc0  = (ocB * 8 + wv) * 32;

  const float* xb = xinT + (size_t)b * HH * WW * IC_;
  float*       ob = outup + (size_t)b * OC_ * UP * UP;

  switch (quad) {   // scalar branch -> EXEC all-ones inside
    case 0: conv_quad<2, 2>(xb, wT, ob, r, 0, 0, oc0, lane); break;
    case 1: conv_quad<2, 1>(xb, wT, ob, r, 0, 1, oc0, lane); break;
    case 2: conv_quad<1, 2>(xb, wT, ob, r, 1, 0, oc0, lane); break;
    default: conv_quad<1, 1>(xb, wT, ob, r, 1, 1, oc0, lane); break;
  }
}

// ---------------- K5: blur(pad(out_up)) * conv_scale * demod ----------------
__global__ __launch_bounds__(256) void k_blur(const float* __restrict__ outup,
                                              const float* __restrict__ demod,
                                              float* __restrict__ out) {
  size_t idx = (size_t)blockIdx.x * 256 + threadIdx.x;
  const size_t total = (size_t)B_ * OC_ * OUTS * OUTS;
  if (idx >= total) return;
  int xx = (int)(idx % OUTS); size_t t = idx / OUTS;
  int yy = (int)(t % OUTS);   t /= OUTS;
  int oc = (int)(t % OC_);
  int b  = (int)(t / OC_);
  const float kv[4] = {1.f, 3.f, 3.f, 1.f};
  const float* up = outup + (size_t)(b * OC_ + oc) * UP * UP;
  float acc = 0.f;
#pragma unroll
  for (int i = 0; i < 4; ++i) {
    int sy = yy + i - 1;
    if (sy < 0 || sy >= UP) continue;
    float rs = 0.f;
#pragma unroll
    for (int j = 0; j < 4; ++j) {
      int sx = xx + j - 1;
      if (sx >= 0 && sx < UP) rs += kv[j] * up[sy * UP + sx];
    }
    acc += kv[i] * rs;
  }
  out[idx] = acc * (1.f / 16.f) * CONV_SCALE * demod[b * OC_ + oc];
}

// ---------------- launch ----------------
extern "C" void kernel_launch(void* const* d_in, const int* in_sizes, int n_in,
                              void* d_out, int out_size, void* d_ws, size_t ws_size,
                              hipStream_t stream) {
  (void)in_sizes; (void)n_in; (void)out_size; (void)ws_size;
  const float* input  = (const float*)d_in[0];   // [16,512,32,32]
  const float* style  = (const float*)d_in[1];   // [16,512]
  const float* weight = (const float*)d_in[2];   // [1,512,512,3,3]
  const float* modw   = (const float*)d_in[3];   // [512,512]
  const float* modb   = (const float*)d_in[4];   // [512]
  float* out = (float*)d_out;                    // [16,512,63,63]

  float* ws    = (float*)d_ws;
  float* s     = ws;                 // 8192
  float* wsq   = s + 8192;           // 262144
  float* demod = wsq + 262144;       // 8192
  float* wT    = demod + 8192;       // 2359296  (9 MB, tap-major K-innermost)
  float* xinT  = wT + 2359296;       // 8388608  (32 MB, NHWC)
  float* outup = xinT + 8388608;     // 33554432 (128 MB)

  k_style<<<32, 256, 0, stream>>>(style, modw, modb, s);
  k_wsq<<<1024, 256, 0, stream>>>(weight, wsq);
  k_demod<<<32, 256, 0, stream>>>(s, wsq, demod);
  k_wtr<<<9216, 256, 0, stream>>>(weight, wT);
  k_modin<<<32768, 256, 0, stream>>>(input, s, xinT);
  k_conv<<<4096, 256, 0, stream>>>(xinT, wT, outup);
  const size_t total = (size_t)B_ * OC_ * OUTS * OUTS;   // 32514048
  k_blur<<<(unsigned)((total + 255) / 256), 256, 0, stream>>>(outup, demod, out);
}